// EmbedSegLossStrict_1202590842953
// MI455X (gfx1250) — compile-verified
//
#include <hip/hip_runtime.h>
#include <hip/hip_bf16.h>
#include <cstdint>

// ---------------- problem constants ----------------
#define BB   4
#define HH   768
#define WW   768
#define NN   (HH * WW)          // 589824 pixels per image
#define KK   16                 // instance ids 1..16
#define MPAD (1 << 20)          // padded segment length for bitonic
#define SEGS_PER_CHUNK 8
#define NCHUNK 8                // 64 segments total
#define EPSF 1e-6f

typedef float v2f __attribute__((ext_vector_type(2)));
typedef float v8f __attribute__((ext_vector_type(8)));

#if defined(__gfx1250__) && __has_builtin(__builtin_amdgcn_wmma_f32_16x16x4_f32)
#define HAVE_WMMA_F32X4 1
#endif

// stats layout: stats[(b*KK+k)*5 + f], f: 0=sum_emb_y 1=sum_emb_x 2=sum_sig_y 3=sum_sig_x 4=count

// ---------------- 0) zero workspace header ----------------
__global__ void zero_kernel(float* wsf) {
    int t = threadIdx.x;
    if (t < 512) wsf[t] = 0.f;
}

// ---------------- 1) per-cluster stats via V_WMMA_F32_16X16X4_F32 ----------------
// D += A(16x4) x B(4x16): A = one-hot mask (16 clusters x 4 pixels),
// B = 4 pixels x 16 features (emb_y, emb_x, sig_y, sig_x, 1, 0...).
// Column 4 (constant 1) accumulates the per-cluster pixel count for free.
// 4x unrolled: 16 pixels / wave iteration, 4 static v_wmma ops.
// Straight-line staging: one row per super-group (768 = 48*16 -> no row crossing),
// hoisted per-lane selector constants -> no EXEC-mask regions in the hot loop.
__global__ void stats_kernel(const float* __restrict__ yp,
                             const int* __restrict__ yt,
                             float* __restrict__ stats) {
    const int b    = blockIdx.y;
    const int tid  = threadIdx.x;
    const int lane = tid & 31;
    const int half = lane >> 4;      // 0: pixels {0,1} / K={0,1};  1: pixels {2,3} / K={2,3}
    const int fm   = lane & 15;      // A row (cluster) == B column (feature)
    const int gw   = blockIdx.x * (blockDim.x >> 5) + (tid >> 5);
    const int nwaves = gridDim.x * (blockDim.x >> 5);
    const int sgroups = NN / 16;         // 36864 super-groups of 16 pixels
    const int GROUPS_PER_ROW = WW / 16;  // 48

    const int*   ytb = yt + (size_t)b * NN;
    const float* ypb = yp + (size_t)b * 5 * NN;
    const int ch = (fm < 4) ? fm : 0;            // branchless channel select
    const float* chp = ypb + (size_t)ch * NN;

    // hoisted per-lane selector constants (loop body becomes pure FMA arithmetic)
    const float gsel_y = (fm == 0) ? 1.f : 0.f;  // add row coordinate
    const float gsel_x = (fm == 1) ? 1.f : 0.f;  // add col coordinate
    const float fsel   = (fm < 4) ? 1.f : 0.f;   // pass loaded channel
    const float onesel = (fm == 4) ? 1.f : 0.f;  // constant-1 (count) column
    const int   target = fm + 1;                 // cluster id for A one-hot

    v8f c = {0.f, 0.f, 0.f, 0.f, 0.f, 0.f, 0.f, 0.f};

    for (int sg = gw; sg < sgroups; sg += nwaves) {
        const int hrow  = sg / GROUPS_PER_ROW;               // row of this 16-pixel group
        const int wbase = (sg - hrow * GROUPS_PER_ROW) << 4; // first col of group
        const float hf  = (float)hrow;
        const float ghy = gsel_y * hf;                       // row term (per group)
        const int   nb  = sg << 4;                           // first pixel index

#pragma unroll
        for (int q = 0; q < 4; ++q) {
            const int o0 = q * 4 + half * 2;
            const int n0 = nb + o0;
            const int n1 = n0 + 1;

            // A: one-hot mask of cluster fm at pixels (n0, n1)
            const int l0 = ytb[n0], l1 = ytb[n1];
            v2f a;
            a.x = (l0 == target) ? 1.f : 0.f;
            a.y = (l1 == target) ? 1.f : 0.f;

            // B: feature fm at pixels (n0, n1) — straight-line arithmetic
            const float v0 = chp[n0];
            const float v1 = chp[n1];
            const float w0 = (float)(wbase + o0);
            const float w1 = (float)(wbase + o0 + 1);
            v2f bv;
            bv.x = fsel * (v0 + ghy + gsel_x * w0) + onesel;
            bv.y = fsel * (v1 + ghy + gsel_x * w1) + onesel;

#if HAVE_WMMA_F32X4
            c = __builtin_amdgcn_wmma_f32_16x16x4_f32(false, a, false, bv,
                                                      (short)0, c, false, false);
#else
            (void)a; (void)bv;   // host pass / missing builtin
#endif
        }
    }

#if HAVE_WMMA_F32X4
    // C layout: lane c<16 vgpr r -> D[M=r][N=c]; lane c+16 -> D[M=r+8][N=c]
    if (fm < 5) {
        for (int r = 0; r < 8; ++r) {
            int k = r + 8 * half;
            atomicAdd(&stats[((size_t)(b * KK + k)) * 5 + fm], c[r]);
        }
    }
#endif
}

// ---------------- 2) seed MSE + variance deviations ----------------
__global__ void seed_var_kernel(const float* __restrict__ yp,
                                const int* __restrict__ yt,
                                const float* __restrict__ stats,
                                float* __restrict__ devsum,
                                float* __restrict__ seed_sum) {
    const int b   = blockIdx.y;
    const int tid = threadIdx.x;
    const int n   = blockIdx.x * blockDim.x + tid;

    __shared__ float sdev[KK];
    __shared__ float sred[256];
    if (tid < KK) sdev[tid] = 0.f;
    __syncthreads();

    float d2 = 0.f;
    if (n < NN) {
        const int*   ytb = yt + (size_t)b * NN;
        const float* ypb = yp + (size_t)b * 5 * NN;
        const int label = ytb[n];
        const float seed = ypb[(size_t)4 * NN + n];
        float target = 0.f;
        if (label > 0) {
            const int k = label - 1;
            const float* st = &stats[((size_t)(b * KK + k)) * 5];
            const float safe = fmaxf(st[4], 1.f);
            const float cky = st[0] / safe, ckx = st[1] / safe;
            const float offy = ypb[n], offx = ypb[(size_t)NN + n];
            const float sgy = ypb[(size_t)2 * NN + n], sgx = ypb[(size_t)3 * NN + n];
            const float ey = (float)(n / WW) + offy;
            const float ex = (float)(n % WW) + offx;
            const float sby = fmaxf(sgy, EPSF), sbx = fmaxf(sgx, EPSF);
            const float dy = ey - cky, dx = ex - ckx;
            const float quad = dy * dy / (2.f * sby * sby) + dx * dx / (2.f * sbx * sbx);
            const float phi = fminf(fmaxf(__expf(-quad), EPSF), 1.f - EPSF);
            target = phi;
            const float msy = st[2] / safe, msx = st[3] / safe;
            atomicAdd(&sdev[k], fabsf(sgy - msy) + fabsf(sgx - msx));
        }
        const float d = seed - target;
        d2 = d * d;
    }
    sred[tid] = d2;
    __syncthreads();
    for (int off = 128; off > 0; off >>= 1) {
        if (tid < off) sred[tid] += sred[tid + off];
        __syncthreads();
    }
    if (tid == 0) atomicAdd(seed_sum, sred[0]);
    if (tid < KK) atomicAdd(&devsum[b * KK + tid], sdev[tid]);
}

// ---------------- 3) build sortable keys for one chunk of 8 segments ----------------
// key u64 = (monotone-f32-bits(err) << 32) | gt ; padding = 0 (sorts to tail, relu(e)=0)
__global__ void fill_keys_kernel(const float* __restrict__ yp,
                                 const int* __restrict__ yt,
                                 const float* __restrict__ stats,
                                 unsigned long long* __restrict__ buf,
                                 int chunk) {
    const int t = blockIdx.x * blockDim.x + threadIdx.x;   // 8 * 2^20 threads
    const int s_local = t >> 20;
    const int n = t & (MPAD - 1);
    const int seg = chunk * SEGS_PER_CHUNK + s_local;
    const int b = seg >> 4, k = seg & 15;

    unsigned long long item = 0ull;
    if (n < NN) {
        const float* st = &stats[(size_t)seg * 5];
        const float safe = fmaxf(st[4], 1.f);
        const float cky = st[0] / safe, ckx = st[1] / safe;
        const float* ypb = yp + (size_t)b * 5 * NN;
        const float offy = ypb[n], offx = ypb[(size_t)NN + n];
        const float sgy = ypb[(size_t)2 * NN + n], sgx = ypb[(size_t)3 * NN + n];
        const float ey = (float)(n / WW) + offy;
        const float ex = (float)(n % WW) + offx;
        const float sby = fmaxf(sgy, EPSF), sbx = fmaxf(sgx, EPSF);
        const float dy = ey - cky, dx = ex - ckx;
        const float quad = dy * dy / (2.f * sby * sby) + dx * dx / (2.f * sbx * sbx);
        const float phi = fminf(fmaxf(__expf(-quad), EPSF), 1.f - EPSF);
        const float logit = __logf(phi) - log1pf(-phi);
        const int gt = (yt[(size_t)b * NN + n] == k + 1) ? 1 : 0;
        const float err = 1.f - logit * (gt ? 1.f : -1.f);
        unsigned int bits = __float_as_uint(err);
        unsigned int u = (bits & 0x80000000u) ? ~bits : (bits | 0x80000000u);
        item = ((unsigned long long)u << 32) | (unsigned long long)gt;
    }
    buf[((size_t)s_local << 20) + n] = item;
}

// ---------------- 4a) bitonic: global compare-exchange pass (j >= 2048) ----------------
__global__ void bitonic_global_kernel(unsigned long long* __restrict__ buf, int j, int kk) {
    const int t = blockIdx.x * blockDim.x + threadIdx.x;   // 8 * 2^20 threads
    const int s = t >> 20;
    const int i = t & (MPAD - 1);
    const int ix = i ^ j;
    if (ix > i) {
        unsigned long long* base = buf + ((size_t)s << 20);
        unsigned long long a = base[i];
        unsigned long long b = base[ix];
        const bool up = ((i & kk) == 0);       // descending overall
        if ((a < b) == up) { base[i] = b; base[ix] = a; }
    }
}

// ---------------- 4b) bitonic: LDS-fused stages (tile = 2048, j <= 1024) ----------------
// Tile staging uses gfx1250 async global->LDS DMA (ASYNCcnt), no VGPR round-trip.
__global__ __launch_bounds__(1024)
void bitonic_lds_kernel(unsigned long long* __restrict__ buf, int kk_param, int mode) {
    __shared__ unsigned long long sh[2048];
    const int tid = threadIdx.x;
    const int tile = blockIdx.x;                 // 4096 tiles (8 segs * 512)
    const int s = tile >> 9;
    const int tile_in_seg = tile & 511;
    unsigned long long* gbase = buf + ((size_t)s << 20) + ((size_t)tile_in_seg << 11);

#if defined(__gfx1250__)
    {
        // LDS aperture keeps the byte offset in addr[31:0]
        unsigned lds0 = (unsigned)(uintptr_t)(&sh[tid]);
        unsigned lds1 = (unsigned)(uintptr_t)(&sh[tid + 1024]);
        unsigned long long g0 = (unsigned long long)(uintptr_t)(gbase + tid);
        unsigned long long g1 = (unsigned long long)(uintptr_t)(gbase + tid + 1024);
        asm volatile("global_load_async_to_lds_b64 %0, %1, off"
                     :: "v"(lds0), "v"(g0) : "memory");
        asm volatile("global_load_async_to_lds_b64 %0, %1, off"
                     :: "v"(lds1), "v"(g1) : "memory");
        asm volatile("s_wait_asynccnt 0x0" ::: "memory");
    }
#else
    sh[tid]        = gbase[tid];
    sh[tid + 1024] = gbase[tid + 1024];
#endif
    __syncthreads();

    const int kk_lo = (mode == 0) ? 2 : kk_param;
    const int kk_hi = (mode == 0) ? 2048 : kk_param;
    for (int kk = kk_lo; kk <= kk_hi; kk <<= 1) {
        const int jstart = (mode == 0) ? (kk >> 1) : 1024;
        for (int j = jstart; j >= 1; j >>= 1) {
            const int i_loc = ((tid & ~(j - 1)) << 1) | (tid & (j - 1));
            const int iseg  = (tile_in_seg << 11) + i_loc;
            const bool up = ((iseg & kk) == 0);
            unsigned long long a = sh[i_loc];
            unsigned long long b = sh[i_loc + j];
            if ((a < b) == up) { sh[i_loc] = b; sh[i_loc + j] = a; }
            __syncthreads();
        }
    }
    gbase[tid]        = sh[tid];
    gbase[tid + 1024] = sh[tid + 1024];
}

// ---------------- 5) Lovász scan per segment (one workgroup each) ----------------
// wave32 ballot/popc scan: 3 barriers per 1024-element tile.
__global__ __launch_bounds__(1024)
void lovasz_reduce_kernel(const unsigned long long* __restrict__ buf,
                          const float* __restrict__ stats,
                          float* __restrict__ iou_sum,
                          int chunk) {
    const int s_local = blockIdx.x;
    const int seg = chunk * SEGS_PER_CHUNK + s_local;
    const float G = stats[(size_t)seg * 5 + 4];
    const unsigned long long* base = buf + ((size_t)s_local << 20);
    const int tid  = threadIdx.x;
    const int lane = tid & 31;
    const int wid  = tid >> 5;

    __shared__ int   wsum[32];
    __shared__ int   wpre[32];
    __shared__ int   stot;
    __shared__ float sred[1024];

    float acc = 0.f;
    int cumBase = 0;

    for (int tile = 0; tile < MPAD / 1024; ++tile) {
        const int pos = tile * 1024 + tid;
        __builtin_prefetch(base + pos + 1024, 0, 0);
        const unsigned long long item = base[pos];
        const int gt = (int)(item & 1ull);
        const unsigned int u = (unsigned int)(item >> 32);
        const unsigned int bits = (u & 0x80000000u) ? (u & 0x7FFFFFFFu) : ~u;
        const float e = __uint_as_float(bits);      // padding -> -NaN -> fmaxf gives 0

        // intra-wave inclusive scan of gt via ballot/popc (wave32)
        const unsigned long long m = __ballot(gt);
        const int incl = (int)__popcll(m & ((1ull << (lane + 1)) - 1ull));
        if (lane == 0) wsum[wid] = (int)__popcll(m);
        __syncthreads();
        if (tid == 0) {
            int run = 0;
#pragma unroll
            for (int w = 0; w < 32; ++w) { wpre[w] = run; run += wsum[w]; }
            stot = run;
        }
        __syncthreads();
        const int cumP = cumBase + wpre[wid] + incl;
        const int tileTot = stot;

        const float fP = (float)cumP;
        const float uni = G + (float)(pos + 1) - fP;
        const float J = 1.f - (G - fP) / fmaxf(uni, 1e-8f);
        float Jm = 0.f;
        if (pos > 0) {
            const float fPm = fP - (float)gt;
            const float um = G + (float)pos - fPm;
            Jm = 1.f - (G - fPm) / fmaxf(um, 1e-8f);
        }
        acc += fmaxf(e, 0.f) * (J - Jm);

        cumBase += tileTot;
        __syncthreads();      // protect wsum/wpre before next tile overwrites
    }

    sred[tid] = acc;
    __syncthreads();
    for (int off = 512; off > 0; off >>= 1) {
        if (tid < off) sred[tid] += sred[tid + off];
        __syncthreads();
    }
    if (tid == 0 && G > 0.f) atomicAdd(iou_sum, sred[0]);
}

// ---------------- 6) combine ----------------
__global__ void final_kernel(const float* __restrict__ stats,
                             const float* __restrict__ devsum,
                             const float* __restrict__ iou_sum,
                             const float* __restrict__ seed_sum,
                             float* __restrict__ out) {
    __shared__ float svar[64];
    __shared__ float spres[64];
    const int t = threadIdx.x;    // 64 threads
    const float cnt = stats[(size_t)t * 5 + 4];
    const float pres = (cnt > 0.f) ? 1.f : 0.f;
    const float safe = fmaxf(cnt, 1.f);
    svar[t]  = devsum[t] / (2.f * safe) * pres;   // D = 2
    spres[t] = pres;
    __syncthreads();
    for (int off = 32; off > 0; off >>= 1) {
        if (t < off) { svar[t] += svar[t + off]; spres[t] += spres[t + off]; }
        __syncthreads();
    }
    if (t == 0) {
        const float np = fmaxf(spres[0], 1.f);
        out[0] = 1.f * (iou_sum[0] / np)
               + 10.f * (svar[0] / np)
               + 1.f * (seed_sum[0] / ((float)BB * (float)NN));
    }
}

// ---------------- host launcher ----------------
extern "C" void kernel_launch(void* const* d_in, const int* in_sizes, int n_in,
                              void* d_out, int out_size, void* d_ws, size_t ws_size,
                              hipStream_t stream) {
    (void)in_sizes; (void)n_in; (void)out_size; (void)ws_size;
    const float* yp = (const float*)d_in[0];
    const int*   yt = (const int*)d_in[1];
    float* out = (float*)d_out;

    float* wsf      = (float*)d_ws;
    float* stats    = wsf;              // 320 floats
    float* devsum   = wsf + 320;        // 64 floats
    float* iou_sum  = wsf + 384;        // 1
    float* seed_sum = wsf + 385;        // 1
    unsigned long long* buf =
        (unsigned long long*)((char*)d_ws + 4096);   // 8 * 2^20 u64 = 64 MB (fits L2)

    zero_kernel<<<1, 512, 0, stream>>>(wsf);
    stats_kernel<<<dim3(64, BB), 256, 0, stream>>>(yp, yt, stats);
    seed_var_kernel<<<dim3((NN + 255) / 256, BB), 256, 0, stream>>>(yp, yt, stats, devsum, seed_sum);

    const int fill_blocks = (SEGS_PER_CHUNK * MPAD) / 256;   // 32768
    for (int chunk = 0; chunk < NCHUNK; ++chunk) {
        fill_keys_kernel<<<fill_blocks, 256, 0, stream>>>(yp, yt, stats, buf, chunk);
        // stages kk = 2..2048 fully inside LDS tiles
        bitonic_lds_kernel<<<4096, 1024, 0, stream>>>(buf, 0, 0);
        for (int kk = 4096; kk <= MPAD; kk <<= 1) {
            for (int j = kk >> 1; j >= 2048; j >>= 1)
                bitonic_global_kernel<<<fill_blocks, 256, 0, stream>>>(buf, j, kk);
            bitonic_lds_kernel<<<4096, 1024, 0, stream>>>(buf, kk, 1);
        }
        lovasz_reduce_kernel<<<SEGS_PER_CHUNK, 1024, 0, stream>>>(buf, stats, iou_sum, chunk);
    }
    final_kernel<<<1, 64, 0, stream>>>(stats, devsum, iou_sum, seed_sum, out);
}